// GNetFVnewGCN_86122684219967
// MI455X (gfx1250) — compile-verified
//
#include <hip/hip_runtime.h>
#include <hip/hip_bf16.h>

// Problem constants (from reference)
#define NNODES   100000
#define NEDGES   1600000
#define IN_CH    128      // 127 features + 1 node_attr
#define HIDDEN   2
#define CH2      (IN_CH * HIDDEN)   // 256
#define OUT_CH   128
#define ATTR     6

typedef __attribute__((ext_vector_type(2))) float v2f;
typedef __attribute__((ext_vector_type(8))) float v8f;

// ---------------------------------------------------------------------------
// Kernel 0: zero the aggregation workspace (N x 256 f32 = 102.4 MB in d_ws)
// ---------------------------------------------------------------------------
__global__ void zero_ws_kernel(float4* __restrict__ p, long n4) {
    long i = (long)blockIdx.x * blockDim.x + threadIdx.x;
    long stride = (long)gridDim.x * blockDim.x;
    float4 z = {0.f, 0.f, 0.f, 0.f};
    for (; i < n4; i += stride) p[i] = z;
}

// ---------------------------------------------------------------------------
// Kernel 1: edge pass. One wave32 per edge (grid-strided).
// Lane l owns channels c = l + 32*j, j=0..3  -> coalesced x_src gathers.
// W_in columns (48 floats) + b_in (8 floats) hoisted into registers: they are
// invariant across edges, so the inner loop is pure FMA + gather + L2 atomics.
// ---------------------------------------------------------------------------
__global__ __launch_bounds__(256) void edge_scatter_kernel(
        const float* __restrict__ x,          // [N,127]
        const float* __restrict__ node_attr,  // [N,1]
        const int*   __restrict__ ei,         // [2,E]
        const float* __restrict__ ea,         // [E,6]
        const float* __restrict__ W_in,       // [6,256]
        const float* __restrict__ b_in,       // [256]
        float*       __restrict__ aggr,       // [N,256]
        int nEdges)
{
    const int lane   = threadIdx.x & 31;
    const int wave   = (int)((blockIdx.x * blockDim.x + threadIdx.x) >> 5);
    const int nWaves = (int)((gridDim.x * blockDim.x) >> 5);

    // Preload per-lane W_in columns / biases (edge-invariant).
    float w[4][2][ATTR];
    float b[4][2];
#pragma unroll
    for (int j = 0; j < 4; ++j) {
        const int c = lane + 32 * j;
#pragma unroll
        for (int h = 0; h < 2; ++h) {
            const int o = c * 2 + h;
            b[j][h] = b_in[o];
#pragma unroll
            for (int k = 0; k < ATTR; ++k)
                w[j][h][k] = W_in[k * CH2 + o];
        }
    }

    for (int e = wave; e < nEdges; e += nWaves) {
        const int src = ei[e];
        const int dst = ei[nEdges + e];

        const float a0 = ea[e * ATTR + 0];
        const float a1 = ea[e * ATTR + 1];
        const float a2 = ea[e * ATTR + 2];
        const float a3 = ea[e * ATTR + 3];
        const float a4 = ea[e * ATTR + 4];
        const float a5 = ea[e * ATTR + 5];

        const float* __restrict__ xrow = x + (long)src * (IN_CH - 1);
        const float  nav  = node_attr[src];
        float* __restrict__ arow = aggr + (long)dst * CH2;

#pragma unroll
        for (int j = 0; j < 4; ++j) {
            const int c = lane + 32 * j;
            const float xv = (c < IN_CH - 1) ? xrow[c] : nav;
#pragma unroll
            for (int h = 0; h < 2; ++h) {
                float s = b[j][h];
                s = fmaf(a0, w[j][h][0], s);
                s = fmaf(a1, w[j][h][1], s);
                s = fmaf(a2, w[j][h][2], s);
                s = fmaf(a3, w[j][h][3], s);
                s = fmaf(a4, w[j][h][4], s);
                s = fmaf(a5, w[j][h][5], s);
                s = fmaxf(s, 0.f);                      // relu
                atomicAdd(&arow[c * 2 + h], s * xv);    // scatter-add (L2 resident)
            }
        }
    }
}

// ---------------------------------------------------------------------------
// Kernel 2: out = tanh(aggr @ W_out + b_out) via V_WMMA_F32_16X16X4_F32.
// Block = 256 threads = 8 waves. blockIdx.x selects a 16-node row tile;
// wave w computes the 16-wide column tile n0 = 16*w (OUT_CH = 8*16).
// A-fragment (16x4 f32): lane<16 holds {K,K+1} of row M=lane, lane>=16 holds
// {K+2,K+3} of row M=lane-16 (ISA 7.12.2). B mirrored. C/D: VGPR r holds
// M=r (lanes 0-15) / M=8+r (lanes 16-31), N = lane&15.
// ---------------------------------------------------------------------------
__global__ __launch_bounds__(256) void node_gemm_tanh_kernel(
        const float* __restrict__ aggr,   // [N,256]
        const float* __restrict__ W_out,  // [256,128]
        const float* __restrict__ b_out,  // [128]
        float*       __restrict__ out,    // [N,128]
        int nNodes)
{
    const int lane = threadIdx.x & 31;
    const int wv   = threadIdx.x >> 5;        // 0..7 -> column tile
    const int m0   = blockIdx.x * 16;
    const int n0   = wv * 16;

    const int half = lane >> 4;               // 0 -> K{0,1}, 1 -> K{2,3}
    const int l    = lane & 15;               // A row / B column within tile

    const float* __restrict__ arow = aggr + (long)(m0 + l) * CH2 + half * 2;
    const float* __restrict__ bcol = W_out + n0 + l;

    v8f acc = {};
#pragma unroll 8
    for (int k = 0; k < CH2; k += 4) {
        v2f a = *(const v2f*)(arow + k);                 // contiguous b64 load
        v2f bb;
        bb.x = bcol[(k + half * 2 + 0) * OUT_CH];
        bb.y = bcol[(k + half * 2 + 1) * OUT_CH];
        acc = __builtin_amdgcn_wmma_f32_16x16x4_f32(
                /*neg_a=*/false, a, /*neg_b=*/false, bb,
                /*c_mod=*/(short)0, acc, /*reuse_a=*/false, /*reuse_b=*/false);
    }

    const float bias = b_out[n0 + l];
#pragma unroll
    for (int r = 0; r < 8; ++r) {
        const int m = m0 + r + half * 8;
        out[(long)m * OUT_CH + n0 + l] = tanhf(acc[r] + bias);
    }
    (void)nNodes;
}

// ---------------------------------------------------------------------------
extern "C" void kernel_launch(void* const* d_in, const int* in_sizes, int n_in,
                              void* d_out, int out_size, void* d_ws, size_t ws_size,
                              hipStream_t stream) {
    const float* x         = (const float*)d_in[0];
    const float* node_attr = (const float*)d_in[1];
    const int*   ei        = (const int*)  d_in[2];
    const float* ea        = (const float*)d_in[3];
    const float* W_in      = (const float*)d_in[4];
    const float* b_in      = (const float*)d_in[5];
    const float* W_out     = (const float*)d_in[6];
    const float* b_out     = (const float*)d_in[7];
    float*       out       = (float*)d_out;
    float*       aggr      = (float*)d_ws;   // needs N*256*4 = 102.4 MB

    const int nNodes = in_sizes[1];          // node_attr is [N,1]
    const int nEdges = in_sizes[3] / ATTR;   // edge_attr is [E,6]

    // 0) zero aggregation workspace
    {
        long n4 = ((long)nNodes * CH2) / 4;
        int blocks = (int)((n4 + 255) / 256);
        zero_ws_kernel<<<blocks, 256, 0, stream>>>((float4*)aggr, n4);
    }
    // 1) edge pass: fused edge-MLP + relu + broadcast-mul + scatter-add
    {
        edge_scatter_kernel<<<4096, 256, 0, stream>>>(
            x, node_attr, ei, ea, W_in, b_in, aggr, nEdges);
    }
    // 2) node pass: fp32 WMMA GEMM + bias + tanh  (N is a multiple of 16)
    {
        int blocks = nNodes / 16;
        node_gemm_tanh_kernel<<<blocks, 256, 0, stream>>>(
            aggr, W_out, b_out, out, nNodes);
    }
}